// approxNDCGLoss_5085241278629
// MI455X (gfx1250) — compile-verified
//
#include <hip/hip_runtime.h>
#include <hip/hip_bf16.h>

// ApproxNDCG loss, B=32, N=2048.
// Key simplifications (see analysis): no argsort needed; maxDCG via 5-bin
// label histogram; pairwise sigmoid sums are permutation invariant.
// Inner loop reduction uses V_WMMA_F32_16X16X4_F32 (exact f32 K=4 matmul
// against an all-ones B) to row-reduce 16x4 sigmoid tiles.

#define NDCG_N 2048
#define NDCG_B 32
#define EPS_F 5e-4f
#define LOG2E_F 1.4426950408889634f
#define ROWS_PER_BLOCK 128               // 8 waves * 16 i-rows
#define CHUNKS (NDCG_N / ROWS_PER_BLOCK) // 16

typedef __attribute__((ext_vector_type(2))) float v2f;
typedef __attribute__((ext_vector_type(8))) float v8f;

__device__ __forceinline__ float block_reduce_256(float v, float* s_red) {
  const int tid = threadIdx.x;
  s_red[tid] = v;
  __syncthreads();
#pragma unroll
  for (int off = 128; off > 0; off >>= 1) {
    if (tid < off) s_red[tid] += s_red[tid + off];
    __syncthreads();
  }
  return s_red[0];
}

// ---------------------------------------------------------------------------
// Kernel 1: per-row maxDCG via label histogram. grid = B, block = 256.
// ---------------------------------------------------------------------------
__global__ void ndcg_stats_kernel(const float* __restrict__ y_true,
                                  float* __restrict__ maxdcg) {
  __shared__ int hist[5];
  __shared__ float s_red[256];
  const int b = blockIdx.x;
  const int tid = threadIdx.x;

  if (tid < 5) hist[tid] = 0;
  __syncthreads();

  for (int t = tid; t < NDCG_N; t += 256) {
    int l = (int)y_true[b * NDCG_N + t];
    l = l < 0 ? 0 : (l > 4 ? 4 : l);
    atomicAdd(&hist[l], 1);
  }
  __syncthreads();

  const int T4 = hist[4];
  const int T3 = T4 + hist[3];
  const int T2 = T3 + hist[2];
  const int T1 = T2 + hist[1];

  // maxDCG = sum over ranks r (0-based) of (2^label(r) - 1) / log2(r + 2);
  // label 0 contributes 0, so loop only to T1 (uniform bound).
  float acc = 0.0f;
  for (int r = tid; r < T1; r += 256) {
    float gain = (r < T4) ? 15.0f : (r < T3) ? 7.0f : (r < T2) ? 3.0f : 1.0f;
    acc += gain / __builtin_amdgcn_logf((float)(r + 2)); // v_log_f32 = log2
  }
  float total = block_reduce_256(acc, s_red);
  if (tid == 0) maxdcg[b] = fmaxf(total, EPS_F);
}

// ---------------------------------------------------------------------------
// Kernel 2: pairwise sigmoid sums + per-element NDCG contributions.
// grid = (CHUNKS, B), block = 256 (8 waves; each wave owns 16 i-rows).
// ---------------------------------------------------------------------------
__global__ void ndcg_pair_kernel(const float* __restrict__ y_pred,
                                 const float* __restrict__ y_true,
                                 const float* __restrict__ maxdcg,
                                 float* __restrict__ partials) {
  __shared__ float s_pred[NDCG_N];   // 8 KB: whole score row
  __shared__ float s_S[ROWS_PER_BLOCK];
  __shared__ float s_red[256];

  const int b = blockIdx.y;
  const int chunk = blockIdx.x;
  const int i_base = chunk * ROWS_PER_BLOCK;
  const int tid = threadIdx.x;

  // Stage the full row of scores into LDS (vectorized b128).
  {
    const float4* src = (const float4*)(y_pred + (size_t)b * NDCG_N);
    float4* dst = (float4*)s_pred;
#pragma unroll
    for (int t = tid; t < NDCG_N / 4; t += 256) dst[t] = src[t];
  }
  __syncthreads();

  const int wave = tid >> 5;
  const int lane = tid & 31;
  const int m = lane & 15;                 // A-matrix row M owned by this lane
  const int koff = (lane < 16) ? 0 : 2;    // A-matrix K columns owned
  const float si = s_pred[i_base + wave * 16 + m];

  v8f c = {0.0f, 0.0f, 0.0f, 0.0f, 0.0f, 0.0f, 0.0f, 0.0f};
#if __has_builtin(__builtin_amdgcn_wmma_f32_16x16x4_f32)
  v2f bones;
  bones.x = 1.0f;
  bones.y = 1.0f;
#else
  float fall_acc = 0.0f;
#endif

#pragma unroll 4
  for (int j0 = 0; j0 < NDCG_N; j0 += 4) {
    // lane reads the two adjacent s_j it owns (8B-aligned LDS b64 load)
    const float2 sj = *reinterpret_cast<const float2*>(&s_pred[j0 + koff]);
    // sig = sigmoid(-(s_i - s_j)) = 1 / (1 + exp(s_i - s_j)); clip at EPS
    const float e0 = __builtin_amdgcn_exp2f((si - sj.x) * LOG2E_F);
    const float e1 = __builtin_amdgcn_exp2f((si - sj.y) * LOG2E_F);
    v2f a;
    a.x = fmaxf(EPS_F, __builtin_amdgcn_rcpf(1.0f + e0));
    a.y = fmaxf(EPS_F, __builtin_amdgcn_rcpf(1.0f + e1));
#if __has_builtin(__builtin_amdgcn_wmma_f32_16x16x4_f32)
    // D[m][n] += sum_k A[m][k] * 1  -> per-row running sum in every column
    c = __builtin_amdgcn_wmma_f32_16x16x4_f32(
        /*neg_a=*/false, a, /*neg_b=*/false, bones,
        /*c_mod=*/(short)0, c, /*reuse_a=*/false, /*reuse_b=*/false);
#else
    fall_acc += a.x + a.y;
#endif
  }

#if __has_builtin(__builtin_amdgcn_wmma_f32_16x16x4_f32)
  // Column N=0 lives in lane 0 (rows 0-7 in c[0..7]) and lane 16 (rows 8-15).
  if (lane == 0) {
#pragma unroll
    for (int r = 0; r < 8; ++r) s_S[wave * 16 + r] = c[r];
  } else if (lane == 16) {
#pragma unroll
    for (int r = 0; r < 8; ++r) s_S[wave * 16 + 8 + r] = c[r];
  }
  __syncthreads();
#else
  if (lane < 16) s_S[wave * 16 + m] = fall_acc;
  __syncthreads();
  if (lane >= 16) s_S[wave * 16 + m] += fall_acc;
  __syncthreads();
#endif

  // Per-element contribution G_i / log2(1 + approx_pos_i).
  float contrib = 0.0f;
  if (tid < ROWS_PER_BLOCK) {
    const int ii = i_base + tid;
    const float S = s_S[tid];                 // includes diagonal term 0.5
    const float approx_pos = 0.5f + S;        // 1 + (S - 0.5)
    const float aD = __builtin_amdgcn_logf(1.0f + approx_pos); // log2
    float g = fmaxf(y_true[(size_t)b * NDCG_N + ii], 0.0f);
    const float G = (__builtin_amdgcn_exp2f(g) - 1.0f) / maxdcg[b];
    contrib = G / aD;
  }
  const float blk = block_reduce_256(contrib, s_red);
  if (tid == 0) partials[b * CHUNKS + chunk] = blk;
}

// ---------------------------------------------------------------------------
// Kernel 3: deterministic final reduce of B*CHUNKS = 512 partials -> mean.
// ---------------------------------------------------------------------------
__global__ void ndcg_final_kernel(const float* __restrict__ partials,
                                  float* __restrict__ out) {
  __shared__ float s_red[256];
  const int tid = threadIdx.x;
  float v = partials[tid] + partials[tid + 256];
  const float total = block_reduce_256(v, s_red);
  if (tid == 0) out[0] = total * (1.0f / (float)NDCG_B);
}

extern "C" void kernel_launch(void* const* d_in, const int* in_sizes, int n_in,
                              void* d_out, int out_size, void* d_ws, size_t ws_size,
                              hipStream_t stream) {
  const float* y_pred = (const float*)d_in[0];
  const float* y_true = (const float*)d_in[1];
  float* out = (float*)d_out;

  float* maxdcg = (float*)d_ws;        // 32 floats
  float* partials = maxdcg + NDCG_B;   // 512 floats

  ndcg_stats_kernel<<<NDCG_B, 256, 0, stream>>>(y_true, maxdcg);
  ndcg_pair_kernel<<<dim3(CHUNKS, NDCG_B), 256, 0, stream>>>(y_pred, y_true,
                                                             maxdcg, partials);
  ndcg_final_kernel<<<1, 256, 0, stream>>>(partials, out);
}